// GAT_82231443849283
// MI455X (gfx1250) — compile-verified
//
#include <hip/hip_runtime.h>

typedef float v2f __attribute__((ext_vector_type(2)));
typedef float v8f __attribute__((ext_vector_type(8)));

#define NEG_SLOPE 0.2f
#define EPS_DEN   1e-16f

// ---- ordered-uint encoding so float max can use atomicMax(u32) ----
__device__ __forceinline__ unsigned enc_ord(float f) {
  unsigned u = __float_as_uint(f);
  return (u & 0x80000000u) ? ~u : (u | 0x80000000u);
}
__device__ __forceinline__ float dec_ord(unsigned u) {
  return (u & 0x80000000u) ? __uint_as_float(u & 0x7FFFFFFFu)
                           : __uint_as_float(~u);
}
#define ENC_NEG_INF 0x007FFFFFu  // enc_ord(-inf)

// ---------------- vectorized workspace fill ----------------
__global__ void k_fill_u32v4(unsigned* __restrict__ p, unsigned v, size_t n) {
  size_t i = ((size_t)blockIdx.x * blockDim.x + threadIdx.x) * 4;
  if (i + 3 < n) {
    uint4 t = make_uint4(v, v, v, v);
    *(uint4*)(p + i) = t;
  } else {
    for (; i < n; ++i) p[i] = v;
  }
}

// ---------------- small-matrix transpose: WT[n,k] = W[k,n] ----------------
__global__ void k_transpose(const float* __restrict__ W, float* __restrict__ WT,
                            int K, int N) {
  int idx = blockIdx.x * blockDim.x + threadIdx.x;
  if (idx >= K * N) return;
  int k = idx / N, n = idx - k * N;
  WT[(size_t)n * K + k] = W[idx];
}

// ---- WMMA GEMM: C[M,N] = A[M,K] x BT[N,K]^T, fp32, K%4==0, N%16==0 ----
// B passed TRANSPOSED so both fragments are contiguous b64 vector loads.
// One wave handles MT row-tiles x one 16-col tile (B fragments reused MT times).
// blockDim.x = 32*(N/16); grid.x = ceil(M/(16*MT)).
// f32 16x16x4 fragment layout (ISA 7.12.2):
//   A: lanes 0-15 hold M=lane, v0=K0 v1=K1; lanes 16-31 hold v0=K2 v1=K3
//   B: mirrored with N=lane%16
//   C/D: v[j] = D[M=j+8*half][N=lane%16]
template <int MT>
__global__ void k_gemm_wmma_f32(const float* __restrict__ A,
                                const float* __restrict__ BT,
                                float* __restrict__ C,
                                int M, int K, int N) {
  const int wave = threadIdx.x >> 5;
  const int lane = threadIdx.x & 31;
  const int r    = lane & 15;
  const int half = lane >> 4;
  const int row0 = blockIdx.x * (16 * MT);
  const int col0 = wave * 16;

  const float* Ap[MT];
#pragma unroll
  for (int m = 0; m < MT; ++m) {
    int arow = row0 + 16 * m + r;
    if (arow >= M) arow = M - 1;               // clamp (tail safety)
    Ap[m] = A + (size_t)arow * K;
  }
  const float* Bp = BT + (size_t)(col0 + r) * K;

  v8f acc[MT];
#pragma unroll
  for (int m = 0; m < MT; ++m) acc[m] = (v8f){};

#pragma unroll 4
  for (int k = 0; k < K; k += 4) {
    const int kk = k + 2 * half;
    v2f b = *(const v2f*)(Bp + kk);            // b64 vector load
#pragma unroll
    for (int m = 0; m < MT; ++m) {
      v2f a = *(const v2f*)(Ap[m] + kk);       // b64 vector load
      acc[m] = __builtin_amdgcn_wmma_f32_16x16x4_f32(false, a, false, b,
                                                     (short)0, acc[m],
                                                     false, false);
    }
  }

#pragma unroll
  for (int m = 0; m < MT; ++m) {
    float* Cp = C + (size_t)(row0 + 16 * m) * N + col0 + r;
#pragma unroll
    for (int j = 0; j < 8; ++j) {
      int row = row0 + 16 * m + j + 8 * half;
      if (row < M) Cp[(size_t)(j + 8 * half) * N] = acc[m][j];
    }
  }
}

// ---- attention dots: as[n,h] = h[n,h,:].att_s[h,:], ad likewise ----
__global__ void k_att_dots(const float* __restrict__ h,
                           const float* __restrict__ att_s,
                           const float* __restrict__ att_d,
                           float* __restrict__ as_, float* __restrict__ ad_,
                           int Nn, int H, int Cc) {
  int idx = blockIdx.x * blockDim.x + threadIdx.x;
  if (idx >= Nn * H) return;
  int nd = idx / H, hh = idx - nd * H;
  const float* hp = h + (size_t)nd * H * Cc + (size_t)hh * Cc;
  const float* sp = att_s + hh * Cc;
  const float* dp = att_d + hh * Cc;
  float accs = 0.f, accd = 0.f;
  for (int c = 0; c < Cc; ++c) { float v = hp[c]; accs += v * sp[c]; accd += v * dp[c]; }
  as_[idx] = accs; ad_[idx] = accd;
}

// ---- segment max over incoming edges (per dst, per head) ----
template <int H>
__global__ void k_edge_max(const long long* __restrict__ ei,
                           const float* __restrict__ as_,
                           const float* __restrict__ ad_,
                           unsigned* __restrict__ emax, long long E) {
  unsigned long long idx = (unsigned long long)blockIdx.x * blockDim.x + threadIdx.x;
  if (idx >= (unsigned long long)E * H) return;
  unsigned long long e = idx / H;              // H power-of-2 -> shift
  int hh = (int)(idx - e * H);
  long long s = ei[e], d = ei[E + e];
  float v = as_[s * H + hh] + ad_[d * H + hh];
  v = v > 0.f ? v : NEG_SLOPE * v;
  atomicMax(&emax[d * H + hh], enc_ord(v));
}

// ---- segment sum of exp(e - max) ----
template <int H>
__global__ void k_edge_expsum(const long long* __restrict__ ei,
                              const float* __restrict__ as_,
                              const float* __restrict__ ad_,
                              const unsigned* __restrict__ emax,
                              float* __restrict__ denom, long long E) {
  unsigned long long idx = (unsigned long long)blockIdx.x * blockDim.x + threadIdx.x;
  if (idx >= (unsigned long long)E * H) return;
  unsigned long long e = idx / H;
  int hh = (int)(idx - e * H);
  long long s = ei[e], d = ei[E + e];
  float v = as_[s * H + hh] + ad_[d * H + hh];
  v = v > 0.f ? v : NEG_SLOPE * v;
  float m = dec_ord(emax[d * H + hh]);
  if (m < -3.0e38f) m = 0.f;                   // isolated-node guard
  atomicAdd(&denom[d * H + hh], __expf(v - m));
}

// ---- aggregate: out[dst] += alpha * h[src]; one wave per edge ----
// Lanes 0..H-1 compute alpha for head=lane; everyone else gets it via shfl.
// Feature gather is a single b128/b64 vector load per lane.
template <int F, int H>
__global__ void k_edge_aggregate(const long long* __restrict__ ei,
                                 const float* __restrict__ h,
                                 const float* __restrict__ as_,
                                 const float* __restrict__ ad_,
                                 const unsigned* __restrict__ emax,
                                 const float* __restrict__ denom,
                                 float* __restrict__ outacc, long long E) {
  constexpr int VPL = F / 32;                  // values per lane (4 or 2)
  constexpr int C   = F / H;
  long long wid = ((long long)blockIdx.x * blockDim.x + threadIdx.x) >> 5;
  int lane = threadIdx.x & 31;
  if (wid >= E) return;
  long long s = ei[wid], d = ei[E + wid];

  float aval = 0.f;
  if (lane < H) {
    float v = as_[s * H + lane] + ad_[d * H + lane];
    v = v > 0.f ? v : NEG_SLOPE * v;
    float m = dec_ord(emax[d * H + lane]);
    if (m < -3.0e38f) m = 0.f;
    aval = __expf(v - m) / (denom[d * H + lane] + EPS_DEN);
  }
  const int c0 = lane * VPL;
  const int hh = c0 / C;                       // compile-time shift
  float alpha = __shfl(aval, hh, 32);

  const float* hp = h + (size_t)s * F + c0;
  float* op = outacc + (size_t)d * F + c0;
  float vals[VPL];
  if constexpr (VPL == 4) {
    float4 t = *(const float4*)hp;             // global_load_b128
    vals[0] = t.x; vals[1] = t.y; vals[2] = t.z; vals[3] = t.w;
  } else {
    float2 t = *(const float2*)hp;             // global_load_b64
    vals[0] = t.x; vals[1] = t.y;
  }
#pragma unroll
  for (int j = 0; j < VPL; ++j) atomicAdd(&op[j], alpha * vals[j]);
}

// ---- in-place bias + ELU (F is power of two) ----
__global__ void k_bias_elu(float* __restrict__ hx, const float* __restrict__ bias,
                           size_t total, int fmask) {
  size_t i = (size_t)blockIdx.x * blockDim.x + threadIdx.x;
  if (i >= total) return;
  float v = hx[i] + bias[i & (size_t)fmask];
  hx[i] = v > 0.f ? v : expm1f(v);
}

// ---- final: out = acc + bias (heads=1, concat=False => mean over 1 head) ----
__global__ void k_bias_out(const float* __restrict__ acc, const float* __restrict__ bias,
                           float* __restrict__ out, size_t total, int fmask) {
  size_t i = (size_t)blockIdx.x * blockDim.x + threadIdx.x;
  if (i >= total) return;
  out[i] = acc[i] + bias[i & (size_t)fmask];
}

static inline unsigned cdiv(size_t a, size_t b) { return (unsigned)((a + b - 1) / b); }

extern "C" void kernel_launch(void* const* d_in, const int* in_sizes, int n_in,
                              void* d_out, int out_size, void* d_ws, size_t ws_size,
                              hipStream_t stream) {
  const float*     x    = (const float*)d_in[0];
  const long long* ei   = (const long long*)d_in[1];   // int64 [2, E]
  const float*     W1   = (const float*)d_in[2];       // [128, 128]
  const float*     as1v = (const float*)d_in[3];       // [8, 16]
  const float*     ad1v = (const float*)d_in[4];
  const float*     b1   = (const float*)d_in[5];       // [128]
  const float*     W2   = (const float*)d_in[6];       // [128, 64]
  const float*     as2v = (const float*)d_in[7];       // [1, 64]
  const float*     ad2v = (const float*)d_in[8];
  const float*     b2   = (const float*)d_in[9];       // [64]
  float*           out  = (float*)d_out;

  const size_t    N = (size_t)in_sizes[0] / 128;
  const long long E = (long long)in_sizes[1] / 2;

  // workspace layout (floats); layer-2 regions recycle layer-1 regions
  float*    ws    = (float*)d_ws;
  float*    h1    = ws;                       // [N,128] (later: h2 [N,64] | aggr2 [N,64])
  float*    as1   = h1 + N * 128;             // [N,8]
  float*    ad1   = as1 + N * 8;              // [N,8]
  unsigned* emax1 = (unsigned*)(ad1 + N * 8); // [N,8]
  float*    den1  = (float*)(emax1 + N * 8);  // [N,8]
  float*    hx    = den1 + N * 8;             // [N,128] aggregated+ELU'd hidden
  float*    wt1   = hx + N * 128;             // [128,128] W1^T
  float*    wt2   = wt1 + 128 * 128;          // [64,128]  W2^T
  float*    h2    = h1;                       // [N,64]
  float*    aggr2 = h1 + N * 64;              // [N,64]
  float*    as2   = as1;                      // [N]
  float*    ad2   = ad1;                      // [N]
  unsigned* emax2 = emax1;                    // [N]
  float*    den2  = den1;                     // [N]

  const int TB = 256;

  // ---------------- weight transposes (tiny; once per launch) ----------------
  k_transpose<<<cdiv(128 * 128, TB), TB, 0, stream>>>(W1, wt1, 128, 128);
  k_transpose<<<cdiv(128 * 64, TB), TB, 0, stream>>>(W2, wt2, 128, 64);

  // ---------------- layer 1 ----------------
  k_fill_u32v4<<<cdiv(N * 8, TB * 4), TB, 0, stream>>>((unsigned*)den1, 0u, N * 8);
  k_fill_u32v4<<<cdiv(N * 8, TB * 4), TB, 0, stream>>>(emax1, ENC_NEG_INF, N * 8);
  k_fill_u32v4<<<cdiv(N * 128, TB * 4), TB, 0, stream>>>((unsigned*)hx, 0u, N * 128);

  // h1 = x @ W1   (M=N, K=128, Nc=128 -> 8 waves/block, 2 row-tiles/wave)
  k_gemm_wmma_f32<2><<<cdiv(N, 32), 8 * 32, 0, stream>>>(x, wt1, h1, (int)N, 128, 128);

  k_att_dots<<<cdiv(N * 8, TB), TB, 0, stream>>>(h1, as1v, ad1v, as1, ad1, (int)N, 8, 16);

  k_edge_max<8><<<cdiv((size_t)E * 8, TB), TB, 0, stream>>>(ei, as1, ad1, emax1, E);
  k_edge_expsum<8><<<cdiv((size_t)E * 8, TB), TB, 0, stream>>>(ei, as1, ad1, emax1, den1, E);
  // one wave per edge, 8 waves per block
  k_edge_aggregate<128, 8><<<cdiv((size_t)E, 8), TB, 0, stream>>>(
      ei, h1, as1, ad1, emax1, den1, hx, E);

  k_bias_elu<<<cdiv(N * 128, TB), TB, 0, stream>>>(hx, b1, N * 128, 127);

  // ---------------- layer 2 ----------------
  k_fill_u32v4<<<cdiv(N, TB * 4), TB, 0, stream>>>((unsigned*)den2, 0u, N);
  k_fill_u32v4<<<cdiv(N, TB * 4), TB, 0, stream>>>(emax2, ENC_NEG_INF, N);
  k_fill_u32v4<<<cdiv(N * 64, TB * 4), TB, 0, stream>>>((unsigned*)aggr2, 0u, N * 64);

  // h2 = hx @ W2   (K=128, Nc=64 -> 4 waves/block, 2 row-tiles/wave)
  k_gemm_wmma_f32<2><<<cdiv(N, 32), 4 * 32, 0, stream>>>(hx, wt2, h2, (int)N, 128, 64);

  k_att_dots<<<cdiv(N, TB), TB, 0, stream>>>(h2, as2v, ad2v, as2, ad2, (int)N, 1, 64);

  k_edge_max<1><<<cdiv((size_t)E, TB), TB, 0, stream>>>(ei, as2, ad2, emax2, E);
  k_edge_expsum<1><<<cdiv((size_t)E, TB), TB, 0, stream>>>(ei, as2, ad2, emax2, den2, E);
  k_edge_aggregate<64, 1><<<cdiv((size_t)E, 8), TB, 0, stream>>>(
      ei, h2, as2, ad2, emax2, den2, aggr2, E);

  k_bias_out<<<cdiv(N * 64, TB), TB, 0, stream>>>(aggr2, b2, out, N * 64, 63);
}